// EventSampler_80564996539201
// MI455X (gfx1250) — compile-verified
//
#include <hip/hip_runtime.h>

// Problem dims (fixed by the reference)
constexpr int B  = 16;
constexpr int S  = 128;
constexpr int NS = 32;
constexpr int NE = 256;
constexpr int ROWS = B * S * NS;          // 65536 rows of NE=256 elements
constexpr int WAVES_PER_BLOCK = 8;        // 8 wave32s per block
constexpr int BLOCK = 32 * WAVES_PER_BLOCK;
#define DTIME_MAX 5.0f

// ---- CDNA5 async global->LDS copy (ASYNCcnt path) ----------------------
// Each lane copies 16 bytes from `g` into LDS byte offset `lds_off`.
// VFLAT async encoding: vdst = LDS byte-address VGPR, vaddr = 64-bit pair.
__device__ __forceinline__ void async_copy_b128(const float* g, unsigned lds_off) {
    asm volatile("global_load_async_to_lds_b128 %0, %1, off"
                 :: "v"(lds_off), "v"(g)
                 : "memory");
}

__device__ __forceinline__ void wait_asynccnt0() {
#if __has_builtin(__builtin_amdgcn_s_wait_asynccnt)
    __builtin_amdgcn_s_wait_asynccnt(0);
#else
    asm volatile("s_wait_asynccnt 0" ::: "memory");
#endif
}

__device__ __forceinline__ unsigned lds_offset_of(const void* p) {
    // Generic LDS pointer = {aperture tag in [63:32], LDS byte offset in [31:0]}
    return (unsigned)(unsigned long long)(uintptr_t)p;
}

__global__ __launch_bounds__(BLOCK)
void EventSampler_80564996539201_kernel(const float* __restrict__ unif,
                                        const float* __restrict__ rate,
                                        const float* __restrict__ tot,
                                        const float* __restrict__ expn,
                                        float* __restrict__ out) {
    // Per-wave staging: 128 floats of unif + 128 floats of tot = 1 KB/wave
    __shared__ float smem[WAVES_PER_BLOCK][256];

    const int wave = threadIdx.x >> 5;
    const int lane = threadIdx.x & 31;
    const int row  = blockIdx.x * WAVES_PER_BLOCK + wave;   // < 65536

    const float r   = rate[row >> 5];       // row / NS -> flat [B,S] index
    const int  base = row * NE;             // fits in int (16.7M)

    float* su = &smem[wave][0];             // staged unif chunk
    float* st = &smem[wave][128];           // staged tot  chunk
    const unsigned su_off = lds_offset_of(su + lane * 4);
    const unsigned st_off = lds_offset_of(st + lane * 4);

    int found = -1;
    #pragma unroll 1
    for (int chunk = 0; chunk < 2; ++chunk) {
        const int cbase = base + chunk * 128;
        // Stage this 128-element chunk of both tensors via the async DMA path.
        async_copy_b128(unif + cbase + lane * 4, su_off);
        async_copy_b128(tot  + cbase + lane * 4, st_off);
        wait_asynccnt0();

        const float4 u = *(const float4*)(su + lane * 4);
        const float4 t = *(const float4*)(st + lane * 4);

        // Exact reference rounding: (u*r)/t < 1.0 with IEEE mul + div.
        // Scan high->low so the smallest accepted offset wins.
        int lo = 4;
        if ((u.w * r) / t.w < 1.0f) lo = 3;
        if ((u.z * r) / t.z < 1.0f) lo = 2;
        if ((u.y * r) / t.y < 1.0f) lo = 1;
        if ((u.x * r) / t.x < 1.0f) lo = 0;

#if __has_builtin(__builtin_amdgcn_ballot_w32)
        unsigned mask = __builtin_amdgcn_ballot_w32(lo < 4);
#else
        unsigned mask = (unsigned)__ballot(lo < 4);
#endif
        if (mask) {
            const int w  = __ffs(mask) - 1;                      // lowest hit lane
            const int lw = __builtin_amdgcn_ds_bpermute(w << 2, lo);
            found = chunk * 128 + (w << 2) + lw;                 // wave-uniform
            break;
        }
    }

    if (lane == 0) {
        out[row] = (found >= 0) ? expn[base + found] : DTIME_MAX;
    }
}

extern "C" void kernel_launch(void* const* d_in, const int* in_sizes, int n_in,
                              void* d_out, int out_size, void* d_ws, size_t ws_size,
                              hipStream_t stream) {
    const float* unif = (const float*)d_in[0];   // [B,S,NS,NE]
    const float* rate = (const float*)d_in[1];   // [B,S]
    const float* tot  = (const float*)d_in[2];   // [B,S,NS,NE]
    const float* expn = (const float*)d_in[3];   // [B,S,NS,NE]
    float* out = (float*)d_out;                  // [B,S,NS]

    const dim3 grid(ROWS / WAVES_PER_BLOCK);     // 8192 blocks
    EventSampler_80564996539201_kernel<<<grid, BLOCK, 0, stream>>>(
        unif, rate, tot, expn, out);
}